// NMS_cpu_2370821948167
// MI455X (gfx1250) — compile-verified
//
#include <hip/hip_runtime.h>
#include <stdint.h>

#define K_NN 64
#define MAX_M_BITS 150001                      // M + 1 (sentinel index M is legal)
#define BITSET_WORDS ((MAX_M_BITS + 31) / 32)  // 4688 words = 18.3 KB LDS
#define RING 8                                 // async pipeline depth (2 KB LDS)

// ---- gfx1250 async global->LDS path (ASYNCcnt), guarded so we compile either way ----
#if defined(__gfx1250__) &&                                             \
    __has_builtin(__builtin_amdgcn_global_load_async_to_lds_b64) &&     \
    __has_builtin(__builtin_amdgcn_s_wait_asynccnt)
#define NMS_ASYNC_LDS 1
#else
#define NMS_ASYNC_LDS 0
#endif

#if NMS_ASYNC_LDS
// Builtin params (per hipcc diagnostics): AS(1)/AS(3) pointers to 64-bit int
// vectors. Integer round-trip casts into addrspace-qualified pointers are
// accepted; LDS generic pointers carry the LDS byte offset in addr[31:0].
typedef int v2i __attribute__((vector_size(8)));
#define AS1_V2I(p) ((__attribute__((address_space(1))) v2i*)(uintptr_t)(p))
#define AS3_V2I(p) ((__attribute__((address_space(3))) v2i*)(uint32_t)(uintptr_t)(p))
#endif

// ---------------------------------------------------------------------------
// Kernel 1: inherently sequential greedy scan, one wave32.
//  - full 150001-bit kept-mask resident in LDS (gfx1250: 320 KB/WGP)
//  - current mask word kept in a register (bits set in increasing i), flushed
//    to LDS once per 32 nodes -> LDS RMW off the loop-carried chain
//  - 8-deep ring of async global->LDS row copies (ASYNCcnt) hides HBM/L2
//    latency; row i+1's indices are preloaded while row i is decided
//  - K=64 = 2 indices/lane in wave32; one __any() ballot per node
// ---------------------------------------------------------------------------
__global__ __launch_bounds__(32) void nms_serial_scan(
    const int* __restrict__ knn, float* __restrict__ kept_out, int m)
{
    __shared__ unsigned int bitset[BITSET_WORDS];
    __shared__ __align__(16) int stage[RING][K_NN];

    const int lane = threadIdx.x;   // 0..31 (wave32)
    const int l2   = lane * 2;

    for (int w = lane; w < BITSET_WORDS; w += 32) bitset[w] = 0u;
    __syncthreads();

    unsigned curword = 0u;          // live mask word for nodes [i & ~31, i]

    // Decision step for node i given its two indices (uniform control flow).
    auto decide = [&](int i, int idx0, int idx1) {
        unsigned w0 = bitset[idx0 >> 5];          // LDS gather (finalized words)
        unsigned w1 = bitset[idx1 >> 5];
        const int cwi = i >> 5;
        if ((idx0 >> 5) == cwi) w0 = curword;     // live word from register
        if ((idx1 >> 5) == cwi) w1 = curword;     // (idx>i bits are 0 in both)
        const bool sup =
            __any((int)(((w0 >> (idx0 & 31)) | (w1 >> (idx1 & 31))) & 1u));
        curword |= (sup ? 0u : 1u) << (i & 31);
        if (lane == 0) kept_out[i] = sup ? 0.0f : 1.0f;
        if ((i & 31) == 31) {                     // flush finalized word
            if (lane == 0) bitset[cwi] = curword; // DS ops in order per wave
            curword = 0u;
        }
    };

#if NMS_ASYNC_LDS
    // Prime the ring: rows 0..RING-1 (8 bytes = 2 indices per lane).
    const int primed = (m < RING) ? m : RING;
    for (int r = 0; r < primed; ++r)
        __builtin_amdgcn_global_load_async_to_lds_b64(
            AS1_V2I(knn + (size_t)r * K_NN + l2), AS3_V2I(&stage[r][l2]), 0, 0);

    int idx0_cur = 0, idx1_cur = 0;
    if (m > RING) __builtin_amdgcn_s_wait_asynccnt(RING - 1);  // row 0 landed
    else          __builtin_amdgcn_s_wait_asynccnt(0);
    if (m > 0) { idx0_cur = stage[0][l2]; idx1_cur = stage[0][l2 + 1]; }

    const int m_main = (m > RING) ? (m - RING) : 0;
    for (int i = 0; i < m_main; ++i) {
        // Rows i+1..i+RING-1 outstanding; <=RING-2 left => row i+1 landed.
        __builtin_amdgcn_s_wait_asynccnt(RING - 2);
        const int nbuf = (i + 1) & (RING - 1);
        const int idx0_nxt = stage[nbuf][l2];
        const int idx1_nxt = stage[nbuf][l2 + 1];

        // Refill slot i&7 with row i+RING. Safe: iteration i-1's ballot
        // consumed gathers issued after this slot's index load (DS in-order),
        // so its LDS reads completed before this issue.
        __builtin_amdgcn_global_load_async_to_lds_b64(
            AS1_V2I(knn + (size_t)(i + RING) * K_NN + l2),
            AS3_V2I(&stage[i & (RING - 1)][l2]), 0, 0);

        // Warm L2 well ahead of the async ring (global_prefetch_b8).
        if ((i & 7) == 0 && i + 64 < m)
            __builtin_prefetch(knn + (size_t)(i + 64) * K_NN + lane * 16, 0, 1);

        decide(i, idx0_cur, idx1_cur);
        idx0_cur = idx0_nxt; idx1_cur = idx1_nxt;
    }

    __builtin_amdgcn_s_wait_asynccnt(0);          // tail rows all resident
    for (int i = m_main; i < m; ++i) {
        decide(i, idx0_cur, idx1_cur);
        if (i + 1 < m) {
            const int nbuf = (i + 1) & (RING - 1);
            idx0_cur = stage[nbuf][l2];
            idx1_cur = stage[nbuf][l2 + 1];
        }
    }
#else
    for (int i = 0; i < m; ++i) {
        const int idx0 = knn[(size_t)i * K_NN + l2];
        const int idx1 = knn[(size_t)i * K_NN + l2 + 1];
        decide(i, idx0, idx1);
    }
#endif
}

// ---------------------------------------------------------------------------
// Kernel 2: kept_knn[i,k] = kept[i] ? knn[i,k] : m.  Pure bandwidth
// (~77 MB total @ 23.3 TB/s ~ 3.3 us). int4/float4 per thread.
// ---------------------------------------------------------------------------
__global__ __launch_bounds__(256) void nms_expand(
    const int* __restrict__ knn, const float* __restrict__ kept,
    float* __restrict__ out, int m)
{
    const size_t quad  = (size_t)blockIdx.x * blockDim.x + threadIdx.x;
    const size_t e     = quad * 4;
    const size_t total = (size_t)m * K_NN;
    if (e >= total) return;

    const int   node  = (int)(e >> 6);         // e / 64
    const float keepf = kept[node];
    const int4  v     = *(const int4*)(knn + e);

    float4 o;
    if (keepf != 0.0f)
        o = make_float4((float)v.x, (float)v.y, (float)v.z, (float)v.w);
    else
        o = make_float4((float)m, (float)m, (float)m, (float)m);

    *(float4*)(out + e) = o;
}

// ---------------------------------------------------------------------------
extern "C" void kernel_launch(void* const* d_in, const int* in_sizes, int n_in,
                              void* d_out, int out_size, void* d_ws, size_t ws_size,
                              hipStream_t stream)
{
    (void)n_in; (void)d_ws; (void)ws_size; (void)out_size;

    // d_in[0] = nodes (float32, M*3) -- dead: reference output ignores it.
    const int* knn = (const int*)d_in[1];       // int indices, M*K flat
    const int  m   = in_sizes[1] / K_NN;        // 150000

    float* out      = (float*)d_out;
    float* kept     = out;                      // first m floats (0/1)
    float* kept_knn = out + m;                  // m*K floats

    // Sequential greedy scan: one wave, mask bitset resident in LDS.
    nms_serial_scan<<<1, 32, 0, stream>>>(knn, kept, m);

    // Parallel expansion of the second output.
    const size_t total  = (size_t)m * K_NN;
    const size_t quads  = total / 4;            // K=64 -> divisible by 4
    const int    tpb    = 256;
    const int    blocks = (int)((quads + tpb - 1) / tpb);
    nms_expand<<<blocks, tpb, 0, stream>>>(knn, kept, kept_knn, m);
}